// LightTransformerBlock_23347442221594
// MI455X (gfx1250) — compile-verified
//
#include <hip/hip_runtime.h>
#include <math.h>

// Problem dims (fixed by the reference)
#define BB    8
#define SS    2048
#define HHH   128
#define NHEAD 4
#define DHEAD 32
#define ROWS  (BB * SS)   // 16384

typedef __attribute__((ext_vector_type(2))) float v2f;
typedef __attribute__((ext_vector_type(8))) float v8f;

// D(16x16,f32) = A(16x4,f32) * B(4x16,f32) + C   -- CDNA5 wave32 WMMA
static __device__ __forceinline__ v8f wmma4(v2f a, v2f b, v8f c) {
  return __builtin_amdgcn_wmma_f32_16x16x4_f32(
      /*neg_a=*/false, a, /*neg_b=*/false, b,
      /*c_mod=*/(short)0, c, /*reuse_a=*/false, /*reuse_b=*/false);
}

// ---- CDNA5 async global->LDS copy (16B per lane), with portable fallback ----
#if __has_builtin(__builtin_amdgcn_global_load_async_to_lds_b128)
#define HAS_ASYNC_LDS 1
#else
#define HAS_ASYNC_LDS 0
#endif

typedef __attribute__((__vector_size__(16))) int v4i_vs;
typedef __attribute__((address_space(1))) v4i_vs* as1_v4i_p;
typedef __attribute__((address_space(3))) v4i_vs* as3_v4i_p;

static __device__ __forceinline__ void cp16(const float* g, float* l) {
#if HAS_ASYNC_LDS
  __builtin_amdgcn_global_load_async_to_lds_b128(
      (as1_v4i_p)g, (as3_v4i_p)l, 0, 0);
#else
  float4 t = *(const float4*)g;
  *(float4*)l = t;
#endif
}
static __device__ __forceinline__ void wait_tile_cur() {   // 16 newer loads may stay in flight
#if HAS_ASYNC_LDS
  asm volatile("s_wait_asynccnt 0x10" ::: "memory");
#else
  asm volatile("s_wait_dscnt 0x0" ::: "memory");
#endif
}
static __device__ __forceinline__ void wait_tile_all() {
#if HAS_ASYNC_LDS
  asm volatile("s_wait_asynccnt 0x0" ::: "memory");
#else
  asm volatile("s_wait_dscnt 0x0" ::: "memory");
#endif
}

// ---------------------------------------------------------------------------
// LayerNorm: one wave per 128-wide row. Unbiased std (ddof=1), /(std+eps).
// ---------------------------------------------------------------------------
__global__ __launch_bounds__(256) void ln_kernel(const float* __restrict__ src,
                                                 const float* __restrict__ ga,
                                                 const float* __restrict__ gb,
                                                 float* __restrict__ dst) {
  int lane = threadIdx.x & 31;
  int wave = threadIdx.x >> 5;
  int row  = blockIdx.x * 8 + wave;
  const float* p = src + (size_t)row * HHH + lane * 4;
  float4 xv = *(const float4*)p;
  float s  = xv.x + xv.y + xv.z + xv.w;
  float sq = xv.x * xv.x + xv.y * xv.y + xv.z * xv.z + xv.w * xv.w;
#pragma unroll
  for (int m = 16; m >= 1; m >>= 1) {
    s  += __shfl_xor(s,  m, 32);
    sq += __shfl_xor(sq, m, 32);
  }
  float mean = s * (1.0f / HHH);
  float var  = (sq - s * mean) * (1.0f / (HHH - 1));
  var = var < 0.0f ? 0.0f : var;
  float inv = 1.0f / (sqrtf(var) + 1e-6f);
  float4 av = *(const float4*)(ga + lane * 4);
  float4 bv = *(const float4*)(gb + lane * 4);
  float4 o;
  o.x = av.x * (xv.x - mean) * inv + bv.x;
  o.y = av.y * (xv.y - mean) * inv + bv.y;
  o.z = av.z * (xv.z - mean) * inv + bv.z;
  o.w = av.w * (xv.w - mean) * inv + bv.w;
  *(float4*)(dst + (size_t)row * HHH + lane * 4) = o;
}

// ---------------------------------------------------------------------------
// Flash attention (Q=K=V=n1). One wave per 16-query tile per (batch, head).
// 64 keys per iteration; K/V tile double-buffered in LDS via async copies.
// ---------------------------------------------------------------------------
#define ATT_WPB 2
#define KT      64            // keys per iteration
#define KVSTR   36            // padded row stride (floats) of K/V tile
#define PSTR    66            // padded row stride (floats) of P staging

static __device__ __forceinline__ void stage_tile(const float* base, int k0,
                                                  float* buf, int lane) {
  // 64 rows x 32 floats = 512 x 16B chunks, 16 per lane
#pragma unroll
  for (int i = 0; i < 16; ++i) {
    int chunk = i * 32 + lane;
    int key   = chunk >> 3;         // 0..63
    int sub   = (chunk & 7) * 4;    // 0..28 (floats)
    cp16(base + (size_t)(k0 + key) * HHH + sub, buf + key * KVSTR + sub);
  }
}

__global__ __launch_bounds__(ATT_WPB * 32) void attn_kernel(
    const float* __restrict__ n1, const int* __restrict__ mask,
    const float* __restrict__ aw, float* __restrict__ out) {
  __shared__ float kv_all[ATT_WPB][2][KT * KVSTR];
  __shared__ float pst_all[ATT_WPB][16 * PSTR];

  int lane = threadIdx.x & 31;
  int wave = threadIdx.x >> 5;
  float* pst = pst_all[wave];

  int gw    = blockIdx.x * ATT_WPB + wave;
  int qt    = gw & (SS / 16 - 1);       // 128 q-tiles
  int tmp   = gw >> 7;
  int head  = tmp & (NHEAD - 1);
  int batch = tmp >> 2;
  int q0    = qt * 16;
  int ln16  = lane & 15;
  int g     = lane >> 4;                // half-wave group 0/1

  const float* base = n1 + ((size_t)batch * SS) * HHH + head * DHEAD;

  // Q fragments in A-layout: a.x = Q[m][4j + 2g], a.y = Q[m][4j + 2g + 1]
  v2f qa[8];
  {
    const float* qrow = base + (size_t)(q0 + ln16) * HHH;
#pragma unroll
    for (int j = 0; j < 8; ++j) {
      float2 t = *(const float2*)(qrow + 4 * j + 2 * g);
      qa[j].x = t.x; qa[j].y = t.y;
    }
  }

  // Per-query-row scale: attn_scale_w[h, q] / sqrt(DH)
  const float invs = 0.17677669529663687f;  // 1/sqrt(32)
  float rs[8];
#pragma unroll
  for (int r = 0; r < 8; ++r)
    rs[r] = aw[head * SS + q0 + r + 8 * g] * invs;

  float mrow[8], lrow[8];
  v8f o0 = {}; v8f o1 = {};
#pragma unroll
  for (int r = 0; r < 8; ++r) { mrow[r] = -1e30f; lrow[r] = 0.0f; }

  const int* mk = mask + batch * SS;

  stage_tile(base, 0, kv_all[wave][0], lane);   // prefetch tile 0
  int cur = 0;

  for (int k0 = 0; k0 < SS; k0 += KT) {
    const float* kv = kv_all[wave][cur];
    bool lastTile = (k0 + KT >= SS);
    if (!lastTile) {
      stage_tile(base, k0 + KT, kv_all[wave][cur ^ 1], lane);  // overlap next
      wait_tile_cur();
    } else {
      wait_tile_all();
    }

    // ---- scores for 4 sub-tiles of 16 keys (Q K^T via WMMA) ----
    int kmv[4];
#pragma unroll
    for (int t = 0; t < 4; ++t) kmv[t] = mk[k0 + 16 * t + ln16];

    float p[4][8];
#pragma unroll
    for (int t = 0; t < 4; ++t) {
      v8f c = {};
#pragma unroll
      for (int j = 0; j < 8; ++j) {
        float2 kt = *(const float2*)(kv + (16 * t + ln16) * KVSTR + 4 * j + 2 * g);
        v2f kb; kb.x = kt.x; kb.y = kt.y;
        c = wmma4(qa[j], kb, c);
      }
#pragma unroll
      for (int r = 0; r < 8; ++r)
        p[t][r] = (kmv[t] == 0) ? -1e9f : c[r] * rs[r];
    }

    // ---- one online-softmax update per 64 keys ----
#pragma unroll
    for (int r = 0; r < 8; ++r) {
      float v = fmaxf(fmaxf(p[0][r], p[1][r]), fmaxf(p[2][r], p[3][r]));
#pragma unroll
      for (int m = 8; m >= 1; m >>= 1) v = fmaxf(v, __shfl_xor(v, m, 32));
      float mnew = fmaxf(mrow[r], v);
      float corr = __expf(mrow[r] - mnew);
      mrow[r] = mnew;
      float ps = 0.0f;
#pragma unroll
      for (int t = 0; t < 4; ++t) {
        p[t][r] = __expf(p[t][r] - mnew);
        ps += p[t][r];
      }
#pragma unroll
      for (int m = 8; m >= 1; m >>= 1) ps += __shfl_xor(ps, m, 32);
      lrow[r] = lrow[r] * corr + ps;
      o0[r] *= corr;
      o1[r] *= corr;
    }

    // ---- stage P (C-layout -> LDS, A-layout reload) ----
    asm volatile("" ::: "memory");
#pragma unroll
    for (int t = 0; t < 4; ++t)
#pragma unroll
      for (int r = 0; r < 8; ++r)
        pst[(r + 8 * g) * PSTR + 16 * t + ln16] = p[t][r];
    asm volatile("s_wait_dscnt 0x0" ::: "memory");

    // ---- O += P V (K dim = 64 keys, two 16-col output tiles) ----
#pragma unroll
    for (int kk = 0; kk < KT; kk += 4) {
      float2 at = *(const float2*)(pst + ln16 * PSTR + kk + 2 * g);
      v2f a; a.x = at.x; a.y = at.y;
      const float* vr = kv + (size_t)(kk + 2 * g) * KVSTR + ln16;
      v2f b0, b1;
      b0.x = vr[0];       b0.y = vr[KVSTR];
      b1.x = vr[16];      b1.y = vr[16 + KVSTR];
      o0 = wmma4(a, b0, o0);
      o1 = wmma4(a, b1, o1);
    }
    asm volatile("" ::: "memory");
    cur ^= 1;
  }

  // ---- normalize by l and write h[b, q, head*32 + d] ----
  float* orow = out + ((size_t)batch * SS) * HHH + head * DHEAD;
#pragma unroll
  for (int r = 0; r < 8; ++r) {
    float invl = 1.0f / lrow[r];
    int m = r + 8 * g;
    float* dst = orow + (size_t)(q0 + m) * HHH;
    dst[ln16]      = o0[r] * invl;
    dst[16 + ln16] = o1[r] * invl;
  }
}

// ---------------------------------------------------------------------------
// GEMM: out = resid + epilogue(A @ W^T + bias). One wave per 16x128 strip.
// GELU path: hf = rowscale[s] * (.); act = 0.5*hf*(1+tanh(0.79788456*(hf+0.044715*hf^3)))
// ---------------------------------------------------------------------------
template <bool GELU>
__global__ __launch_bounds__(128) void gemm_kernel(
    const float* __restrict__ A, const float* __restrict__ W,
    const float* __restrict__ bias, const float* __restrict__ resid,
    const float* __restrict__ rowscale, float* __restrict__ out) {
  int lane = threadIdx.x & 31;
  int wave = threadIdx.x >> 5;
  int gw   = blockIdx.x * 4 + wave;
  int row0 = gw * 16;
  int ln16 = lane & 15;
  int g    = lane >> 4;

  v8f acc[8] = {};
  const float* arow = A + (size_t)(row0 + ln16) * HHH;
#pragma unroll 4
  for (int kk = 0; kk < HHH; kk += 4) {
    float2 at = *(const float2*)(arow + kk + 2 * g);
    v2f a; a.x = at.x; a.y = at.y;
#pragma unroll
    for (int nt = 0; nt < 8; ++nt) {
      int col = nt * 16 + ln16;
      float2 wt = *(const float2*)(W + (size_t)col * HHH + kk + 2 * g);
      v2f bf; bf.x = wt.x; bf.y = wt.y;
      acc[nt] = wmma4(a, bf, acc[nt]);  // B[k][n] = W[n][k]  (W^T)
    }
  }

#pragma unroll
  for (int nt = 0; nt < 8; ++nt) {
    int col = nt * 16 + ln16;
    float bv = bias[col];
#pragma unroll
    for (int r = 0; r < 8; ++r) {
      int row = row0 + r + 8 * g;
      float val = acc[nt][r] + bv;
      if constexpr (GELU) {
        float sc = rowscale[row & (SS - 1)];   // per-sequence-position scale
        float hf = sc * val;
        float t = tanhf(0.7978845608028654f * (hf + 0.044715f * hf * hf * hf));
        val = 0.5f * hf * (1.0f + t);
      }
      out[(size_t)row * HHH + col] = resid[(size_t)row * HHH + col] + val;
    }
  }
}

// ---------------------------------------------------------------------------
extern "C" void kernel_launch(void* const* d_in, const int* in_sizes, int n_in,
                              void* d_out, int out_size, void* d_ws, size_t ws_size,
                              hipStream_t stream) {
  const float* hidden       = (const float*)d_in[0];
  const int*   mask         = (const int*)d_in[1];
  const float* attn_scale_w = (const float*)d_in[2];
  const float* Wl           = (const float*)d_in[3];
  const float* bl           = (const float*)d_in[4];
  const float* W1           = (const float*)d_in[5];
  const float* b1           = (const float*)d_in[6];
  const float* ffn_scale_w  = (const float*)d_in[7];
  const float* ln1_a        = (const float*)d_in[8];
  const float* ln1_b        = (const float*)d_in[9];
  const float* ln2_a        = (const float*)d_in[10];
  const float* ln2_b        = (const float*)d_in[11];

  const size_t NFLT = (size_t)ROWS * HHH;  // 2,097,152 floats
  float* n1 = (float*)d_ws;
  float* ah = n1 + NFLT;
  float* x2 = ah + NFLT;
  float* n2 = x2 + NFLT;
  float* outp = (float*)d_out;

  // 1) n1 = LN1(x)
  ln_kernel<<<ROWS / 8, 256, 0, stream>>>(hidden, ln1_a, ln1_b, n1);
  // 2) ah = attention(n1)  (flash, WMMA f32, async LDS double-buffering)
  attn_kernel<<<(BB * NHEAD * (SS / 16)) / ATT_WPB, ATT_WPB * 32, 0, stream>>>(
      n1, mask, attn_scale_w, ah);
  // 3) x2 = x + ah @ Wl^T + bl
  gemm_kernel<false><<<(ROWS / 16) / 4, 128, 0, stream>>>(ah, Wl, bl, hidden,
                                                          nullptr, x2);
  // 4) n2 = LN2(x2)
  ln_kernel<<<ROWS / 8, 256, 0, stream>>>(x2, ln2_a, ln2_b, n2);
  // 5) out = x2 + gelu(ffn_scale * (n2 @ W1^T + b1))
  gemm_kernel<true><<<(ROWS / 16) / 4, 128, 0, stream>>>(n2, W1, b1, x2,
                                                         ffn_scale_w, outp);
}